// GATEncoder_15453292331383
// MI455X (gfx1250) — compile-verified
//
#include <hip/hip_runtime.h>
#include <math.h>

#define N_NODES  100000
#define N_EDGES  400000
#define E_TOT    (N_EDGES + N_NODES)   // self-loops appended
#define N_GRAPHS 2000
#define IN_DIM   58
#define K1PAD    64                    // layer-1 K padded to multiple of 4
#define HC       256                   // heads*ch for every layer (4*64 or 1*256)
#define NEG_SLOPE 0.2f

typedef __attribute__((ext_vector_type(2))) float v2f;
typedef __attribute__((ext_vector_type(8))) float v8f;

__device__ __forceinline__ void atomicMaxF(float* addr, float val) {
    unsigned int* ua = reinterpret_cast<unsigned int*>(addr);
    unsigned int cur = __float_as_uint(*addr);
    while (__uint_as_float(cur) < val) {
        unsigned int prev = atomicCAS(ua, cur, __float_as_uint(val));
        if (prev == cur) break;
        cur = prev;
    }
}

// ---------------------------------------------------------------------------
// Dense GEMM  C[M,HC] = A[M,K] @ B[K,HC]  via V_WMMA_F32_16X16X4_F32.
// K is compile-time (64 or 256) -> no bounds checks, unconditional b64 loads.
// One wave owns a 16x64 output strip (4 N-tiles): one A fragment per k-step
// feeds 4 WMMAs.  8 waves / 256-thread block.
// f32 A 16x4 layout: lanes 0-15 hold {K=0,K=1}, lanes 16-31 hold {K=2,K=3},
// M = lane&15.  B mirrored with N = lane&15.  C/D: VGPR r -> M = r (+8 hi).
// ---------------------------------------------------------------------------
template<int K>
__global__ void wmma_gemm_f32(const float* __restrict__ A,
                              const float* __restrict__ B,
                              float* __restrict__ C, int M) {
    constexpr int NC = HC;
    constexpr int NT = 4;                       // N-tiles per wave
    int wave = blockIdx.x * (blockDim.x >> 5) + (threadIdx.x >> 5);
    int lane = threadIdx.x & 31;
    int groupsN = NC / (16 * NT);               // 4
    int totalGroups = (M >> 4) * groupsN;
    if (wave >= totalGroups) return;
    int tM = wave / groupsN;
    int g  = wave - tM * groupsN;
    bool hi = lane >= 16;
    int  ml = lane & 15;
    int  row  = tM * 16 + ml;                   // A row for this lane
    int  col0 = g * (16 * NT) + ml;             // first B/C column for this lane

    const v2f* Arow = reinterpret_cast<const v2f*>(A + (size_t)row * K);

    v8f acc[NT] = {};
    for (int k0 = 0; k0 < K; k0 += 4) {
        int ka = k0 + (hi ? 2 : 0);
        v2f a = Arow[ka >> 1];                  // {A[row,ka], A[row,ka+1]}
        const float* Br0 = B + (size_t)ka * NC + col0;
        const float* Br1 = Br0 + NC;
#pragma unroll
        for (int t = 0; t < NT; ++t) {
            v2f b;
            b.x = Br0[t * 16];
            b.y = Br1[t * 16];
            acc[t] = __builtin_amdgcn_wmma_f32_16x16x4_f32(
                false, a, false, b, (short)0, acc[t], false, false);
        }
    }
#pragma unroll
    for (int r = 0; r < 8; ++r) {
        int mrow = tM * 16 + (hi ? 8 + r : r);
        float* Crow = C + (size_t)mrow * NC + col0;
#pragma unroll
        for (int t = 0; t < NT; ++t) Crow[t * 16] = acc[t][r];
    }
}

// zero-pad a row-major [M,Ksrc] matrix into [M,Kdst] (Kdst >= Ksrc)
__global__ void pad_cols(const float* __restrict__ src, float* __restrict__ dst,
                         int M, int Ksrc, int Kdst) {
    int i = blockIdx.x * blockDim.x + threadIdx.x;
    if (i >= M * Kdst) return;
    int r = i / Kdst, c = i - r * Kdst;
    dst[i] = (c < Ksrc) ? src[(size_t)r * Ksrc + c] : 0.0f;
}

// zero-pad rows: [Rsrc,HC] -> [Rdst,HC]
__global__ void pad_rows(const float* __restrict__ src, float* __restrict__ dst,
                         int Rsrc, int Rdst) {
    int i = blockIdx.x * blockDim.x + threadIdx.x;
    if (i >= Rdst * HC) return;
    dst[i] = (i < Rsrc * HC) ? src[i] : 0.0f;
}

// es[n,h] = <h[n,h,:], a_src[h,:]> ; ed likewise.  One thread per (n,h).
__global__ void attn_scores(const float* __restrict__ h,
                            const float* __restrict__ a_src,
                            const float* __restrict__ a_dst,
                            float* __restrict__ es, float* __restrict__ ed,
                            int H, int C) {
    int idx = blockIdx.x * blockDim.x + threadIdx.x;
    if (idx >= N_NODES * H) return;
    int n  = idx / H;
    int hh = idx - n * H;
    const float* hp = h + (size_t)n * H * C + (size_t)hh * C;
    const float* as = a_src + (size_t)hh * C;
    const float* ad = a_dst + (size_t)hh * C;
    float s = 0.0f, d = 0.0f;
    for (int c = 0; c < C; ++c) { float v = hp[c]; s += v * as[c]; d += v * ad[c]; }
    es[idx] = s;
    ed[idx] = d;
}

__global__ void fill_f32(float* __restrict__ p, float v, int n) {
    int i = blockIdx.x * blockDim.x + threadIdx.x;
    if (i < n) p[i] = v;
}

// pass 1: e = leakyrelu(es[src]+ed[dst]); eval <- e; segment max into m[dst,h]
__global__ void edge_max(const int* __restrict__ ei,
                         const float* __restrict__ es, const float* __restrict__ ed,
                         float* __restrict__ eval, float* __restrict__ m, int H) {
    int idx = blockIdx.x * blockDim.x + threadIdx.x;
    if (idx >= E_TOT * H) return;
    int e  = idx / H;
    int hh = idx - e * H;
    int s, d;
    if (e < N_EDGES) { s = ei[e]; d = ei[N_EDGES + e]; } else { s = d = e - N_EDGES; }
    float v = es[s * H + hh] + ed[d * H + hh];
    v = (v > 0.0f) ? v : NEG_SLOPE * v;
    eval[idx] = v;
    atomicMaxF(&m[d * H + hh], v);
}

// pass 2: eval <- exp(e - m[dst]); denom[dst,h] += eval
__global__ void edge_expsum(const int* __restrict__ ei,
                            const float* __restrict__ m,
                            float* __restrict__ eval, float* __restrict__ denom, int H) {
    int idx = blockIdx.x * blockDim.x + threadIdx.x;
    if (idx >= E_TOT * H) return;
    int e  = idx / H;
    int hh = idx - e * H;
    int d  = (e < N_EDGES) ? ei[N_EDGES + e] : (e - N_EDGES);
    float ex = __expf(eval[idx] - m[d * H + hh]);
    eval[idx] = ex;
    atomicAdd(&denom[d * H + hh], ex);
}

// pass 3: out[dst, h*C+c] += h[src, h*C+c] * eval/denom.  One block per edge.
__global__ void edge_agg(const int* __restrict__ ei,
                         const float* __restrict__ h,
                         const float* __restrict__ eval,
                         const float* __restrict__ denom,
                         float* __restrict__ out, int H, int C) {
    int e = blockIdx.x;
    int c = threadIdx.x;                      // 0..HC-1
    int s, d;
    if (e < N_EDGES) { s = ei[e]; d = ei[N_EDGES + e]; } else { s = d = e - N_EDGES; }
    int hh = c / C;
    float coef = eval[e * H + hh] / denom[d * H + hh];
    atomicAdd(&out[(size_t)d * HC + c], h[(size_t)s * HC + c] * coef);
}

// out += bias, optional ReLU, in place
__global__ void bias_act(float* __restrict__ buf, const float* __restrict__ b,
                         int n, int relu) {
    int i = blockIdx.x * blockDim.x + threadIdx.x;
    if (i >= n) return;
    float v = buf[i] + b[i % HC];
    if (relu) v = v > 0.0f ? v : 0.0f;
    buf[i] = v;
}

// global max pool: out[batch[n], c] = max(..., h[n,c])
__global__ void pool_max(const float* __restrict__ h, const int* __restrict__ batch,
                         float* __restrict__ out) {
    int i = blockIdx.x * blockDim.x + threadIdx.x;
    if (i >= N_NODES * HC) return;
    int n = i / HC;
    int c = i - n * HC;
    atomicMaxF(&out[batch[n] * HC + c], h[i]);
}

static inline int cdiv(long long a, long long b) { return (int)((a + b - 1) / b); }

extern "C" void kernel_launch(void* const* d_in, const int* in_sizes, int n_in,
                              void* d_out, int out_size, void* d_ws, size_t ws_size,
                              hipStream_t stream) {
    const float* x        = (const float*)d_in[0];
    const int*   ei       = (const int*)d_in[1];     // [2, N_EDGES] flat
    const int*   batch    = (const int*)d_in[2];
    const float* W1       = (const float*)d_in[3];
    const float* a_src1   = (const float*)d_in[4];
    const float* a_dst1   = (const float*)d_in[5];
    const float* b1       = (const float*)d_in[6];
    const float* W2       = (const float*)d_in[7];
    const float* a_src2   = (const float*)d_in[8];
    const float* a_dst2   = (const float*)d_in[9];
    const float* b2       = (const float*)d_in[10];
    const float* W3       = (const float*)d_in[11];
    const float* a_src3   = (const float*)d_in[12];
    const float* a_dst3   = (const float*)d_in[13];
    const float* b3       = (const float*)d_in[14];
    float* out = (float*)d_out;

    // ---- workspace carve (256B aligned) ----
    char* p = (char*)d_ws;
    auto carve = [&](size_t bytes) {
        void* r = (void*)p;
        p += (bytes + 255) & ~(size_t)255;
        return r;
    };
    float* bufA  = (float*)carve((size_t)N_NODES * HC * 4);    // GEMM output features
    float* bufB  = (float*)carve((size_t)N_NODES * HC * 4);    // aggregation accumulator
    float* xpad  = (float*)carve((size_t)N_NODES * K1PAD * 4); // x padded 58->64
    float* w1pad = (float*)carve((size_t)K1PAD * HC * 4);      // W1 padded 58->64 rows
    float* es    = (float*)carve((size_t)N_NODES * 4 * 4);
    float* ed    = (float*)carve((size_t)N_NODES * 4 * 4);
    float* mmax  = (float*)carve((size_t)N_NODES * 4 * 4);
    float* denom = (float*)carve((size_t)N_NODES * 4 * 4);
    float* eval  = (float*)carve((size_t)E_TOT   * 4 * 4);

    const int TB = 256;
    const float NEG_INF = -3.402823466e38f;
    const int gemmGroups = (N_NODES / 16) * (HC / 64);         // wave = 16x64 strip
    const int gemmBlocks = cdiv(gemmGroups, TB / 32);

    // pad layer-1 operands once
    pad_cols<<<cdiv((long long)N_NODES * K1PAD, TB), TB, 0, stream>>>(x, xpad, N_NODES, IN_DIM, K1PAD);
    pad_rows<<<cdiv((long long)K1PAD * HC, TB), TB, 0, stream>>>(W1, w1pad, IN_DIM, K1PAD);

    auto gat_tail = [&](const float* asrc, const float* adst, const float* bias,
                        int H, int C, int relu) {
        attn_scores<<<cdiv((long long)N_NODES * H, TB), TB, 0, stream>>>(bufA, asrc, adst, es, ed, H, C);
        fill_f32<<<cdiv((long long)N_NODES * H, TB), TB, 0, stream>>>(mmax, NEG_INF, N_NODES * H);
        fill_f32<<<cdiv((long long)N_NODES * H, TB), TB, 0, stream>>>(denom, 0.0f, N_NODES * H);
        fill_f32<<<cdiv((long long)N_NODES * HC, TB), TB, 0, stream>>>(bufB, 0.0f, N_NODES * HC);
        edge_max   <<<cdiv((long long)E_TOT * H, TB), TB, 0, stream>>>(ei, es, ed, eval, mmax, H);
        edge_expsum<<<cdiv((long long)E_TOT * H, TB), TB, 0, stream>>>(ei, mmax, eval, denom, H);
        edge_agg   <<<E_TOT, HC, 0, stream>>>(ei, bufA, eval, denom, bufB, H, C);
        bias_act   <<<cdiv((long long)N_NODES * HC, TB), TB, 0, stream>>>(bufB, bias, N_NODES * HC, relu);
    };

    // layer 1: 58(->64) -> 4x64 concat, ReLU
    wmma_gemm_f32<K1PAD><<<gemmBlocks, TB, 0, stream>>>(xpad, w1pad, bufA, N_NODES);
    gat_tail(a_src1, a_dst1, b1, 4, 64, 1);
    // layer 2: 256 -> 4x64 concat, ReLU   (bufB from layer 1 is the input)
    wmma_gemm_f32<HC><<<gemmBlocks, TB, 0, stream>>>(bufB, W2, bufA, N_NODES);
    gat_tail(a_src2, a_dst2, b2, 4, 64, 1);
    // layer 3: 256 -> 1x256, no ReLU
    wmma_gemm_f32<HC><<<gemmBlocks, TB, 0, stream>>>(bufB, W3, bufA, N_NODES);
    gat_tail(a_src3, a_dst3, b3, 1, 256, 0);

    // global max pool over graphs
    fill_f32<<<cdiv((long long)N_GRAPHS * HC, TB), TB, 0, stream>>>(out, NEG_INF, N_GRAPHS * HC);
    pool_max<<<cdiv((long long)N_NODES * HC, TB), TB, 0, stream>>>(bufB, batch, out);
}